// WeightedTensorProductBlock2_10144712753192
// MI455X (gfx1250) — compile-verified
//
#include <hip/hip_runtime.h>

// WeightedTensorProductBlock2 for MI455X (gfx1250, wave32, WMMA)
//
// out1[m=(n,i,j)][kf=(a,b)] = x[n,i,a]*y[n,j,b]            (302 MB stream store)
// out2[m][k] = 0.75*a * sum_kf out1[m][kf] * w[kf][k]       (GEMM 4608x16384x128)
//
// Fused: each WG owns a 32-row M-tile. Phase 1 streams the out1 tile from
// LDS-resident x/y rows (coalesced float4). Phase 2 runs the GEMM with
// V_WMMA_F32_16X16X4_F32, regenerating A on the fly (rank-1 structure) so out1
// is never re-read from HBM; each wave keeps TWO 16x16 accumulators sharing
// every B load. w is staged through LDS in 32x128 chunks (L2-resident, 8 MB).

typedef __attribute__((ext_vector_type(2))) float v2f;
typedef __attribute__((ext_vector_type(8))) float v8f;
typedef __attribute__((ext_vector_type(4))) float f4;

#define NB    512
#define DIN   128                 // IN1 == IN2 == KOUT == 128
#define MTOT  (NB * 9)            // 4608 rows (n,i,j)
#define KFLAT (DIN * DIN)         // 16384 contraction length (a,b)
#define NORMC 0.75f
#define MT    32                  // M-tile rows per workgroup
#define KC    32                  // w rows staged in LDS per chunk (divides 128)

__global__ __launch_bounds__(256)
void wtp2_fused_kernel(const float* __restrict__ w,
                       const float* __restrict__ a_scalar,
                       const float* __restrict__ x,
                       const float* __restrict__ y,
                       float* __restrict__ out1,
                       float* __restrict__ out2)
{
    __shared__ __align__(16) float xs[MT][DIN];     // 16 KB
    __shared__ __align__(16) float ys[MT][DIN];     // 16 KB
    __shared__ __align__(16) float ws[KC][DIN];     // 16 KB: w[kc..kc+31][0..127]

    const int tid  = threadIdx.x;        // 0..255 (8 wave32)
    const int lane = tid & 31;
    const int wv   = tid >> 5;           // wave id 0..7 -> k-columns wv*16..wv*16+15
    const int m0   = blockIdx.x * MT;    // M-tile base row

    // ---------------- stage x/y rows of this 32-row tile into LDS ----------------
    for (int idx = tid; idx < MT * DIN; idx += 256) {
        int r = idx >> 7, c = idx & (DIN - 1);
        int m = m0 + r;
        int n = m / 9, rem = m - n * 9;
        int i = rem / 3, j = rem - i * 3;
        xs[r][c] = x[(size_t)(n * 3 + i) * DIN + c];
        ys[r][c] = y[(size_t)(n * 3 + j) * DIN + c];
    }
    __syncthreads();

    // ---------------- phase 1: stream out1 tile (bandwidth-bound) ----------------
    {
        const int r = tid >> 3;          // 8 threads cooperate on one row
        const int c = tid & 7;
        const float* xr = xs[r];
        const float* yr = ys[r];
        f4* dst = (f4*)(out1 + (size_t)(m0 + r) * KFLAT);
        #pragma unroll 4
        for (int it = 0; it < KFLAT / (4 * 8); ++it) {    // 512 iterations
            int q  = c + (it << 3);      // float4 index within row (coalesced x8)
            int ab = q << 2;
            int a  = ab >> 7;            // feature index of x
            int b  = ab & (DIN - 1);     // feature index of y (4-aligned, no wrap)
            float xv = xr[a];
            f4 v;
            v.x = xv * yr[b + 0];
            v.y = xv * yr[b + 1];
            v.z = xv * yr[b + 2];
            v.w = xv * yr[b + 3];
            dst[q] = v;
        }
    }

    // ---------------- phase 2: GEMM via V_WMMA_F32_16X16X4_F32 ----------------
    // A (16x4 f32 layout): lane<16 holds {K0,K1} of row=lane; lane>=16 holds {K2,K3}.
    // B (4x16 layout, mirrored): rows K0/K2 over lane halves, col = lane&15.
    // Two accumulators per wave: M rows [0..15] and [16..31] of the tile share B.
    v8f acc0 = {};
    v8f acc1 = {};
    const int l15 = lane & 15;
    const int kh  = (lane >> 4) << 1;    // 0 for lanes 0-15, 2 for lanes 16-31
    const int kk  = (wv << 4) + l15;     // global output column k

    for (int kc = 0; kc < KFLAT; kc += KC) {
        __syncthreads();
        // stage 32x128 chunk of w into LDS (fully coalesced b128 loads)
        {
            const f4* src = (const f4*)(w + (size_t)kc * DIN);
            f4* dstl = (f4*)&ws[0][0];
            #pragma unroll
            for (int idx = tid; idx < KC * DIN / 4; idx += 256)
                dstl[idx] = src[idx];
        }
        // prefetch next chunk toward L2 while we compute on this one
        if (kc + KC < KFLAT)
            __builtin_prefetch(w + (size_t)(kc + KC) * DIN + tid * 16, 0, 1);
        __syncthreads();

        const int   ac  = kc >> 7;          // 'a' index (constant over a 32-row chunk)
        const int   b0  = kc & (DIN - 1);   // 'b' base within the a-block
        const float xa0 = xs[l15][ac];      // A row scalars, hoisted per chunk
        const float xa1 = xs[l15 + 16][ac];
        const float* yr0 = ys[l15];
        const float* yr1 = ys[l15 + 16];

#if __has_builtin(__builtin_amdgcn_wmma_f32_16x16x4_f32)
        #pragma unroll
        for (int bb = 0; bb < KC; bb += 4) {
            v2f B;
            B.x = ws[bb + kh + 0][kk];      // shared by both accumulators
            B.y = ws[bb + kh + 1][kk];
            v2f A0, A1;
            A0.x = xa0 * yr0[b0 + bb + kh + 0];
            A0.y = xa0 * yr0[b0 + bb + kh + 1];
            A1.x = xa1 * yr1[b0 + bb + kh + 0];
            A1.y = xa1 * yr1[b0 + bb + kh + 1];
            // 8 args: (neg_a, A, neg_b, B, c_mod, C, reuse_a, reuse_b)
            acc0 = __builtin_amdgcn_wmma_f32_16x16x4_f32(
                false, A0, false, B, (short)0, acc0, false, false);
            acc1 = __builtin_amdgcn_wmma_f32_16x16x4_f32(
                false, A1, false, B, (short)0, acc1, false, false);
        }
#else
        // compile-safety fallback (scalar FMA), same result layout
        {
            const int mh = (lane >> 4) << 3;
            #pragma unroll
            for (int bb = 0; bb < KC; ++bb) {
                float wv2 = ws[bb][kk];
                #pragma unroll
                for (int r = 0; r < 8; ++r) {
                    acc0[r] += xs[mh + r][ac] * ys[mh + r][b0 + bb] * wv2;
                    acc1[r] += xs[mh + r + 16][ac] * ys[mh + r + 16][b0 + bb] * wv2;
                }
            }
        }
#endif
    }

    // ---------------- epilogue: scale by 0.75*a, write out2 ----------------
    // C/D layout: VGPR r -> M = r + (lane>=16 ? 8 : 0), N column = lane&15.
    const float scale = NORMC * a_scalar[0];
    const int mhalf = (lane >> 4) << 3;
    #pragma unroll
    for (int r = 0; r < 8; ++r) {
        int mA = m0 + mhalf + r;            // accumulator 0: tile rows 0..15
        int mB = m0 + 16 + mhalf + r;       // accumulator 1: tile rows 16..31
        out2[(size_t)mA * DIN + kk] = acc0[r] * scale;
        out2[(size_t)mB * DIN + kk] = acc1[r] * scale;
    }
}

extern "C" void kernel_launch(void* const* d_in, const int* in_sizes, int n_in,
                              void* d_out, int out_size, void* d_ws, size_t ws_size,
                              hipStream_t stream) {
    // setup_inputs order: w [128,128,128], a [1], x [512,3,128], y [512,3,128]
    const float* w = (const float*)d_in[0];
    const float* a = (const float*)d_in[1];
    const float* x = (const float*)d_in[2];
    const float* y = (const float*)d_in[3];

    float* out1 = (float*)d_out;                                // 4608*16384 floats
    float* out2 = out1 + (size_t)MTOT * KFLAT;                  // 4608*128 floats

    wtp2_fused_kernel<<<MTOT / MT, 256, 0, stream>>>(w, a, x, y, out1, out2);
}